// GraphExpert_86406152061590
// MI455X (gfx1250) — compile-verified
//
#include <hip/hip_runtime.h>
#include <hip/hip_bf16.h>

#define NN 50000
#define NE 600000
#define NB 4096

typedef __attribute__((ext_vector_type(2))) float v2f;
typedef __attribute__((ext_vector_type(8))) float v8f;

// Build Wcat [IN][3*OUT] = [ root | W0 | W1 ],  W_r = comp[r,0]*bases[0] + comp[r,1]*bases[1]
__global__ __launch_bounds__(256) void build_wcat(const float* __restrict__ comp,
                                                  const float* __restrict__ bases,
                                                  const float* __restrict__ root,
                                                  float* __restrict__ Wcat,
                                                  int IN, int OUT) {
    int ncols = OUT * 3;
    int t = blockIdx.x * blockDim.x + threadIdx.x;
    if (t >= IN * ncols) return;
    int i = t / ncols, cc = t % ncols;
    float v;
    if (cc < OUT) {
        v = root[i * OUT + cc];
    } else {
        int r = cc / OUT - 1, o = cc % OUT;
        v = comp[r * 2 + 0] * bases[(size_t)i * OUT + o]
          + comp[r * 2 + 1] * bases[(size_t)(IN + i) * OUT + o];
    }
    Wcat[t] = v;
}

// Per-(dst, relation) edge counts (shared by both layers).
__global__ __launch_bounds__(256) void edge_count(const int* __restrict__ dst,
                                                  const int* __restrict__ et,
                                                  float* __restrict__ cnt, int E) {
    int e = blockIdx.x * blockDim.x + threadIdx.x;
    if (e >= E) return;
    atomicAdd(&cnt[(size_t)dst[e] * 2 + et[e]], 1.0f);
}

// Y[M x NCOLS] = X[M x 128] @ W[128 x NCOLS], fp32 WMMA 16x16x4.
// One wave per 16-row tile; 4 C accumulators -> 16x64 output per wave.
template <int NCOLS>
__global__ __launch_bounds__(128) void rgcn_gemm(const float* __restrict__ X,
                                                 const float* __restrict__ W,
                                                 float* __restrict__ Y, int M) {
    const int lane = threadIdx.x & 31;
    const int wid  = threadIdx.x >> 5;
    const int rowTile = blockIdx.x * 4 + wid;
    if (rowTile * 16 >= M) return;          // whole-wave uniform: EXEC stays all-ones
    const int m0   = rowTile * 16;
    const int n0   = blockIdx.y * 64;
    const int half = lane >> 4;             // 0: lanes 0-15, 1: lanes 16-31
    const int lrow = lane & 15;

    v8f c0 = {}, c1 = {}, c2 = {}, c3 = {};
    const float* xrow = X + (size_t)(m0 + lrow) * 128;

    for (int k0 = 0; k0 < 128; k0 += 4) {
        const int ka = k0 + half * 2;
        // A 16x4: VGPR0 = {K0 | K2}, VGPR1 = {K1 | K3}  -> contiguous float2 per lane
        v2f a;
        a.x = xrow[ka];
        a.y = xrow[ka + 1];
        // B 4x16: VGPR0 = row K=ka, VGPR1 = row K=ka+1, N striped across 16 lanes
        const float* wr0 = W + (size_t)ka * NCOLS + n0 + lrow;
        const float* wr1 = wr0 + NCOLS;
        v2f b0; b0.x = wr0[0];  b0.y = wr1[0];
        v2f b1; b1.x = wr0[16]; b1.y = wr1[16];
        v2f b2; b2.x = wr0[32]; b2.y = wr1[32];
        v2f b3; b3.x = wr0[48]; b3.y = wr1[48];
        c0 = __builtin_amdgcn_wmma_f32_16x16x4_f32(false, a, false, b0, (short)0, c0, false, false);
        c1 = __builtin_amdgcn_wmma_f32_16x16x4_f32(false, a, false, b1, (short)0, c1, false, false);
        c2 = __builtin_amdgcn_wmma_f32_16x16x4_f32(false, a, false, b2, (short)0, c2, false, false);
        c3 = __builtin_amdgcn_wmma_f32_16x16x4_f32(false, a, false, b3, (short)0, c3, false, false);
    }
    // C/D 16x16 f32: VGPR v -> row v (lanes 0-15) / row v+8 (lanes 16-31), col = lane%16
#pragma unroll
    for (int v = 0; v < 8; ++v) {
        float* yr = Y + (size_t)(m0 + v + 8 * half) * NCOLS + n0 + lrow;
        yr[0]  = c0[v];
        yr[16] = c1[v];
        yr[32] = c2[v];
        yr[48] = c3[v];
    }
}

// Gather Y[src] (relation block) and scatter-add into agg[dst][rel][*]; 4 channels/thread.
__global__ __launch_bounds__(256) void edge_agg(const float* __restrict__ Y,
                                                const int* __restrict__ src,
                                                const int* __restrict__ dst,
                                                const int* __restrict__ et,
                                                float* __restrict__ agg,
                                                int E, int ncols, int ystride, int yoff) {
    const int lanesPerEdge = ncols >> 2;
    long long t = (long long)blockIdx.x * blockDim.x + threadIdx.x;
    long long e = t / lanesPerEdge;
    int li = (int)(t - e * lanesPerEdge);
    if (e >= E) return;
    int s = src[e], d = dst[e], r = et[e];
    const float4 v = *(const float4*)(Y + (size_t)s * ystride + yoff + (size_t)r * ncols + li * 4);
    float* ap = agg + ((size_t)d * 2 + r) * ncols + li * 4;
    atomicAdd(ap + 0, v.x);
    atomicAdd(ap + 1, v.y);
    atomicAdd(ap + 2, v.z);
    atomicAdd(ap + 3, v.w);
}

// out[i,c] = (maybe relu)( Y[i,c] + bias[c] + agg0[i,c]/max(cnt0,1) + agg1[i,c]/max(cnt1,1) )
__global__ __launch_bounds__(256) void combine(const float* __restrict__ Y,
                                               const float* __restrict__ bias,
                                               const float* __restrict__ agg,
                                               const float* __restrict__ cnt,
                                               float* __restrict__ H,
                                               int N, int ncols, int ystride, int doRelu) {
    long long t = (long long)blockIdx.x * blockDim.x + threadIdx.x;
    int i = (int)(t / ncols), c = (int)(t % ncols);
    if (i >= N) return;
    float c0 = fmaxf(cnt[(size_t)i * 2 + 0], 1.0f);
    float c1 = fmaxf(cnt[(size_t)i * 2 + 1], 1.0f);
    float v = Y[(size_t)i * ystride + c] + bias[c]
            + agg[((size_t)i * 2 + 0) * ncols + c] / c0
            + agg[((size_t)i * 2 + 1) * ncols + c] / c1;
    H[(size_t)i * ncols + c] = doRelu ? fmaxf(v, 0.0f) : v;
}

// Gather batch nodes, 64->32 relu, 32->1 sigmoid. One wave per batch element.
__global__ __launch_bounds__(256) void head(const float* __restrict__ H2,
                                            const int* __restrict__ idx,
                                            const float* __restrict__ w1,
                                            const float* __restrict__ b1,
                                            const float* __restrict__ w2,
                                            const float* __restrict__ b2,
                                            float* __restrict__ out, int B) {
    int lane = threadIdx.x & 31;
    int b = (blockIdx.x * blockDim.x + threadIdx.x) >> 5;
    if (b >= B) return;
    const float* er = H2 + (size_t)idx[b] * 64;
    out[(size_t)b * 64 + lane]      = er[lane];
    out[(size_t)b * 64 + 32 + lane] = er[32 + lane];
    float h = b1[lane];                      // lane j computes hid[j], j = 0..31
    for (int c = 0; c < 64; ++c) h += er[c] * w1[c * 32 + lane];
    h = fmaxf(h, 0.0f);
    float p = h * w2[lane];
    for (int off = 16; off > 0; off >>= 1) p += __shfl_down(p, off, 32);
    if (lane == 0) out[(size_t)B * 64 + b] = 1.0f / (1.0f + __expf(-(p + b2[0])));
}

extern "C" void kernel_launch(void* const* d_in, const int* in_sizes, int n_in,
                              void* d_out, int out_size, void* d_ws, size_t ws_size,
                              hipStream_t stream) {
    const float* x      = (const float*)d_in[0];
    const float* comp1  = (const float*)d_in[1];
    const float* bases1 = (const float*)d_in[2];
    const float* root1  = (const float*)d_in[3];
    const float* bias1  = (const float*)d_in[4];
    const float* comp2  = (const float*)d_in[5];
    const float* bases2 = (const float*)d_in[6];
    const float* root2  = (const float*)d_in[7];
    const float* bias2  = (const float*)d_in[8];
    const float* w1     = (const float*)d_in[9];
    const float* b1     = (const float*)d_in[10];
    const float* w2     = (const float*)d_in[11];
    const float* b2     = (const float*)d_in[12];
    const int*   nidx   = (const int*)d_in[13];
    const int*   eidx   = (const int*)d_in[14];
    const int*   etype  = (const int*)d_in[15];
    const int* esrc = eidx;
    const int* edst = eidx + NE;

    float* ws = (float*)d_ws;
    size_t off = 0;
    float* W1c = ws + off; off += 128 * 384;
    float* W2c = ws + off; off += 128 * 192;
    float* cnt = ws + off; off += 100352;              // 2*NN, padded for 16B alignment
    float* Y   = ws + off; off += (size_t)NN * 384;    // GEMM output, both layers
    float* H   = ws + off; off += (size_t)NN * 128;    // layer-1 activations
    float* AGG = ws + off; off += (size_t)NN * 256;    // per-(node,rel) accumulators
    float* H2  = ws + off; off += (size_t)NN * 64;     // layer-2 output

    hipMemsetAsync(cnt, 0, (size_t)100352 * 4, stream);
    hipMemsetAsync(AGG, 0, (size_t)NN * 256 * 4, stream);

    build_wcat<<<(128 * 384 + 255) / 256, 256, 0, stream>>>(comp1, bases1, root1, W1c, 128, 128);
    build_wcat<<<(128 * 192 + 255) / 256, 256, 0, stream>>>(comp2, bases2, root2, W2c, 128, 64);
    edge_count<<<(NE + 255) / 256, 256, 0, stream>>>(edst, etype, cnt, NE);

    // ---- layer 1: X[NN,128] @ [root1|W1_0|W1_1] -> Y[NN,384]
    rgcn_gemm<384><<<dim3((NN / 16 + 3) / 4, 6), 128, 0, stream>>>(x, W1c, Y, NN);
    edge_agg<<<(int)(((long long)NE * 32 + 255) / 256), 256, 0, stream>>>(
        Y, esrc, edst, etype, AGG, NE, 128, 384, 128);
    combine<<<(int)(((long long)NN * 128 + 255) / 256), 256, 0, stream>>>(
        Y, bias1, AGG, cnt, H, NN, 128, 384, 1);

    // ---- layer 2: H[NN,128] @ [root2|W2_0|W2_1] -> Y[NN,192]
    hipMemsetAsync(AGG, 0, (size_t)NN * 128 * 4, stream);
    rgcn_gemm<192><<<dim3((NN / 16 + 3) / 4, 3), 128, 0, stream>>>(H, W2c, Y, NN);
    edge_agg<<<(int)(((long long)NE * 16 + 255) / 256), 256, 0, stream>>>(
        Y, esrc, edst, etype, AGG, NE, 64, 192, 64);
    combine<<<(int)(((long long)NN * 64 + 255) / 256), 256, 0, stream>>>(
        Y, bias2, AGG, cnt, H2, NN, 64, 192, 0);

    // ---- head: gather + MLP
    head<<<(NB * 32 + 255) / 256, 256, 0, stream>>>(H2, nidx, w1, b1, w2, b2, (float*)d_out, NB);
}